// SAGE_4604204941856
// MI455X (gfx1250) — compile-verified
//
#include <hip/hip_runtime.h>

typedef float v2f __attribute__((ext_vector_type(2)));
typedef float v8f __attribute__((ext_vector_type(8)));

// ---------------------------------------------------------------- zero ----
__global__ void sage_zero_kernel(float* __restrict__ p, int n) {
    int i = blockIdx.x * blockDim.x + threadIdx.x;
    if (i < n) p[i] = 0.0f;
}

// ------------------------------------------------- edge scatter (mean) ----
// One wave32 per edge. Each lane moves a float4 chunk of the feature row of
// x[src[e]] into hsum[dst[e]] via f32 global atomics (L2-resident target).
// Lane 0 accumulates the float degree (matches segment_sum of ones).
__global__ void sage_scatter_kernel(const float* __restrict__ x,
                                    const int* __restrict__ src,
                                    const int* __restrict__ dst,
                                    float* __restrict__ hsum,
                                    float* __restrict__ deg,
                                    int E, int F) {
    int gtid = blockIdx.x * blockDim.x + threadIdx.x;
    int edge = gtid >> 5;
    int lane = threadIdx.x & 31;
    if (edge >= E) return;
    int s = src[edge];
    int d = dst[edge];
    const float4* __restrict__ xs = (const float4*)(x + (size_t)s * F);
    float* __restrict__ hd = hsum + (size_t)d * F;
    int nf4 = F >> 2;
    for (int f = lane; f < nf4; f += 32) {
        float4 v = xs[f];
        atomicAdd(hd + 4 * f + 0, v.x);
        atomicAdd(hd + 4 * f + 1, v.y);
        atomicAdd(hd + 4 * f + 2, v.z);
        atomicAdd(hd + 4 * f + 3, v.w);
    }
    if (lane == 0) atomicAdd(deg + d, 1.0f);
}

// ------------------------------------------------------- mean division ----
__global__ void sage_meandiv_kernel(float* __restrict__ hsum,
                                    const float* __restrict__ deg,
                                    int n_dst, int F) {
    int i = blockIdx.x * blockDim.x + threadIdx.x;
    if (i >= n_dst * F) return;
    int node = i / F;
    float dg = fmaxf(deg[node], 1.0f);
    hsum[i] *= (1.0f / dg);
}

// --------------------------------------------------------- WMMA GEMM ------
// out[m,n] = act( sum_k Aself[m,k]  * Wself[k,n]   (k < K1)
//               + sum_k Aneigh[m,k] * Wneigh[k,n]  (k < K2)  + bias[n] )
//
// One wave computes a 64x16 output stripe (4 M-subtiles of 16x16) with
// V_WMMA_F32_16X16X4_F32. All loads are UNCONDITIONAL: out-of-range rows and
// columns are address-CLAMPED (they only affect output rows/cols that are
// never stored), so there is no exec-mask divergence in the hot loop.
// All strides/dims are template constants so every inner-loop address is a
// base register + immediate offset (no s_mul_u64 in the loop).
// Fragment layouts per CDNA5 ISA 7.12.2 (f32 shapes):
//   A 16x4 : lanes 0-15 -> {K=k+0,k+1}, lanes 16-31 -> {K=k+2,k+3}, M=lane&15
//   B 4x16 : VGPR0/1 hold rows k+koff(+1), koff = (lane>>4)*2, N=lane&15
//   C 16x16: VGPR r -> M = (lane>>4)*8 + r, N = lane&15
template <int LDA_S, int LDA_N, int K1, int K2, int LDW, int N, int LDO, int DO_RELU>
__global__ __launch_bounds__(128)
void sage_gemm_wmma_kernel(const float* __restrict__ Aself,
                           const float* __restrict__ Aneigh,
                           const float* __restrict__ Wself,
                           const float* __restrict__ Wneigh,
                           const float* __restrict__ bias,
                           float* __restrict__ out, int M) {
    constexpr int nTilesN = (N + 15) >> 4;
    const int lane  = threadIdx.x;                       // 0..31, wave32
    const int tileN = blockIdx.x * blockDim.y + threadIdx.y;
    if (tileN >= nTilesN) return;                        // wave-uniform exit

    const int m0    = blockIdx.y * 64;                   // 4 subtiles of 16 rows
    const int col   = tileN * 16 + (lane & 15);
    const int colc  = min(col, N - 1);                   // clamped (stores guarded)
    const int koff  = (lane >> 4) * 2;                   // upper half-wave: K+2,K+3
    const int rl    = lane & 15;

    // Row addresses, clamped to M-1 (M is a multiple of 64 here, so no-op).
    const int r0 = min(m0 + rl,      M - 1);
    const int r1 = min(m0 + rl + 16, M - 1);
    const int r2 = min(m0 + rl + 32, M - 1);
    const int r3 = min(m0 + rl + 48, M - 1);

    v8f acc0 = {}, acc1 = {}, acc2 = {}, acc3 = {};

    {   // ---- self term: A = Aself (M x K1), B = Wself (K1 x N) ----
        const float* __restrict__ a0 = Aself + (size_t)r0 * LDA_S + koff;
        const float* __restrict__ a1 = Aself + (size_t)r1 * LDA_S + koff;
        const float* __restrict__ a2 = Aself + (size_t)r2 * LDA_S + koff;
        const float* __restrict__ a3 = Aself + (size_t)r3 * LDA_S + koff;
        const float* __restrict__ w  = Wself + (size_t)koff * LDW + colc;
        #pragma unroll 8
        for (int k = 0; k < K1; k += 4) {
            v2f fa0 = *(const v2f*)(a0 + k);
            v2f fa1 = *(const v2f*)(a1 + k);
            v2f fa2 = *(const v2f*)(a2 + k);
            v2f fa3 = *(const v2f*)(a3 + k);
            v2f fb;
            fb.x = w[k * LDW];
            fb.y = w[k * LDW + LDW];
            acc0 = __builtin_amdgcn_wmma_f32_16x16x4_f32(false, fa0, false, fb, (short)0, acc0, false, false);
            acc1 = __builtin_amdgcn_wmma_f32_16x16x4_f32(false, fa1, false, fb, (short)0, acc1, false, false);
            acc2 = __builtin_amdgcn_wmma_f32_16x16x4_f32(false, fa2, false, fb, (short)0, acc2, false, false);
            acc3 = __builtin_amdgcn_wmma_f32_16x16x4_f32(false, fa3, false, fb, (short)0, acc3, false, false);
        }
    }
    {   // ---- neighbor term: A = Aneigh (M x K2), B = Wneigh (K2 x N) ----
        const float* __restrict__ a0 = Aneigh + (size_t)r0 * LDA_N + koff;
        const float* __restrict__ a1 = Aneigh + (size_t)r1 * LDA_N + koff;
        const float* __restrict__ a2 = Aneigh + (size_t)r2 * LDA_N + koff;
        const float* __restrict__ a3 = Aneigh + (size_t)r3 * LDA_N + koff;
        const float* __restrict__ w  = Wneigh + (size_t)koff * LDW + colc;
        #pragma unroll 8
        for (int k = 0; k < K2; k += 4) {
            v2f fa0 = *(const v2f*)(a0 + k);
            v2f fa1 = *(const v2f*)(a1 + k);
            v2f fa2 = *(const v2f*)(a2 + k);
            v2f fa3 = *(const v2f*)(a3 + k);
            v2f fb;
            fb.x = w[k * LDW];
            fb.y = w[k * LDW + LDW];
            acc0 = __builtin_amdgcn_wmma_f32_16x16x4_f32(false, fa0, false, fb, (short)0, acc0, false, false);
            acc1 = __builtin_amdgcn_wmma_f32_16x16x4_f32(false, fa1, false, fb, (short)0, acc1, false, false);
            acc2 = __builtin_amdgcn_wmma_f32_16x16x4_f32(false, fa2, false, fb, (short)0, acc2, false, false);
            acc3 = __builtin_amdgcn_wmma_f32_16x16x4_f32(false, fa3, false, fb, (short)0, acc3, false, false);
        }
    }

    if (col < N) {
        const float bval = bias[colc];
        const int mrow = (lane >> 4) * 8;     // C layout: upper half-lanes = M+8
        #pragma unroll
        for (int r = 0; r < 8; ++r) {
            float v0 = acc0[r] + bval;
            float v1 = acc1[r] + bval;
            float v2 = acc2[r] + bval;
            float v3 = acc3[r] + bval;
            if (DO_RELU) {
                v0 = fmaxf(v0, 0.0f); v1 = fmaxf(v1, 0.0f);
                v2 = fmaxf(v2, 0.0f); v3 = fmaxf(v3, 0.0f);
            }
            int m = m0 + mrow + r;
            if (m      < M) out[(size_t)(m)      * LDO + col] = v0;
            if (m + 16 < M) out[(size_t)(m + 16) * LDO + col] = v1;
            if (m + 32 < M) out[(size_t)(m + 32) * LDO + col] = v2;
            if (m + 48 < M) out[(size_t)(m + 48) * LDO + col] = v3;
        }
    }
}

// ---------------------------------------------------------------------------
extern "C" void kernel_launch(void* const* d_in, const int* in_sizes, int n_in,
                              void* d_out, int out_size, void* d_ws, size_t ws_size,
                              hipStream_t stream) {
    // Problem constants fixed by the reference (n_dst scalars live on device
    // and cannot be read during graph capture).
    const int N2 = 40000, N3 = 8000;
    constexpr int IN = 128, H = 256, C = 47;

    const float* x      = (const float*)d_in[0];
    const int*   src1   = (const int*)  d_in[1];
    const int*   dst1   = (const int*)  d_in[2];
    const int*   src2   = (const int*)  d_in[3];
    const int*   dst2   = (const int*)  d_in[4];
    const float* Wself1 = (const float*)d_in[7];
    const float* Wneigh1= (const float*)d_in[8];
    const float* b1     = (const float*)d_in[9];
    const float* Wself2 = (const float*)d_in[10];
    const float* Wneigh2= (const float*)d_in[11];
    const float* b2     = (const float*)d_in[12];
    float*       outp   = (float*)d_out;

    const int E1 = in_sizes[1];
    const int E2 = in_sizes[3];

    // Workspace layout (floats). First region must be zeroed every call.
    float* ws     = (float*)d_ws;
    float* hsum1  = ws;                                   // N2*IN
    float* deg1   = hsum1 + (size_t)N2 * IN;              // N2
    float* hsum2  = deg1  + N2;                           // N3*H
    float* deg2   = hsum2 + (size_t)N3 * H;               // N3
    const int nzero = N2 * IN + N2 + N3 * H + N3;
    float* h1     = deg2 + N3;                            // N2*H (fully overwritten)

    // 1) zero accumulators
    sage_zero_kernel<<<(nzero + 255) / 256, 256, 0, stream>>>(ws, nzero);

    // 2) layer-1 scatter mean-sum (one wave per edge, 8 edges/block)
    {
        long long threads = (long long)E1 * 32;
        int blocks = (int)((threads + 255) / 256);
        sage_scatter_kernel<<<blocks, 256, 0, stream>>>(x, src1, dst1,
                                                        hsum1, deg1, E1, IN);
    }
    // 3) divide by degree
    sage_meandiv_kernel<<<(N2 * IN + 255) / 256, 256, 0, stream>>>(hsum1, deg1, N2, IN);

    // 4) layer-1 fused dual GEMM + bias + relu -> h1 (N2 x H)
    {
        dim3 blk(32, 4, 1);
        dim3 grd((H / 16 + 3) / 4, N2 / 64, 1);           // 4 x 625
        sage_gemm_wmma_kernel<IN, IN, IN, IN, H, H, H, 1>
            <<<grd, blk, 0, stream>>>(x, hsum1, Wself1, Wneigh1, b1, h1, N2);
    }

    // 5) layer-2 scatter
    {
        long long threads = (long long)E2 * 32;
        int blocks = (int)((threads + 255) / 256);
        sage_scatter_kernel<<<blocks, 256, 0, stream>>>(h1, src2, dst2,
                                                        hsum2, deg2, E2, H);
    }
    // 6) divide by degree
    sage_meandiv_kernel<<<(N3 * H + 255) / 256, 256, 0, stream>>>(hsum2, deg2, N3, H);

    // 7) layer-2 fused dual GEMM + bias -> out (N3 x C), C=47 padded to 3 tiles
    {
        dim3 blk(32, 4, 1);
        dim3 grd(((C + 15) / 16 + 3) / 4, N3 / 64, 1);    // 1 x 125
        sage_gemm_wmma_kernel<H, H, H, H, C, C, C, 0>
            <<<grd, blk, 0, stream>>>(h1, hsum2, Wself2, Wneigh2, b2, outp, N3);
    }
}